// GCN_layer_90546500534889
// MI455X (gfx1250) — compile-verified
//
#include <hip/hip_runtime.h>

// GCN layer, MI455X (gfx1250, wave32).
//   N=4096 nodes, B=16, FIN=FOUT=128, adj ~1% dense.
// Strategy: sparse aggregation (bandwidth-bound, ~130MB HBM => ~6us) fused with
// a per-node 16x128 @ 128x128 FC done on the f32 WMMA pipe (V_WMMA_F32_16X16X4_F32),
// keeping full f32 precision and never spilling the agg intermediate to HBM.

#define N_NODES 4096
#define BATCH   16
#define FEAT    128
#define TPB     256           // 8 wave32 per block
#define APAD    132           // LDS row stride (floats), padded vs 128 to dodge bank conflicts

typedef __attribute__((ext_vector_type(2))) float v2f;
typedef __attribute__((ext_vector_type(8))) float v8f;

// --- degree = 1 (self loop) + column sums of adj;  dinv = rsqrt(degree) ---

__global__ void gcn_deg_init(float* __restrict__ deg) {
    int m = blockIdx.x * TPB + threadIdx.x;
    deg[m] = 1.0f;                       // identity from (A + I)
}

__global__ void gcn_colsum(const float* __restrict__ adj, float* __restrict__ deg) {
    int m  = blockIdx.x * TPB + threadIdx.x;   // column
    int r0 = blockIdx.y * 256;                 // row chunk
    float s = 0.0f;
    for (int r = 0; r < 256; ++r)              // coalesced: lane-consecutive columns
        s += adj[(size_t)(r0 + r) * N_NODES + m];
    atomicAdd(&deg[m], s);
}

__global__ void gcn_dinv(const float* __restrict__ deg, float* __restrict__ dinv) {
    int m = blockIdx.x * TPB + threadIdx.x;
    dinv[m] = rsqrtf(deg[m]);
}

// --- fused: sparse aggregation (regs) -> LDS 16x128 tile -> WMMA f32 FC -> out ---

__global__ void __launch_bounds__(TPB, 2)
gcn_agg_fc(const float* __restrict__ adj,
           const float* __restrict__ x,      // [B, N, FEAT]
           const float* __restrict__ W,      // [FOUT, FIN]
           const float* __restrict__ bias,   // [FOUT]
           const float* __restrict__ dinv,   // [N]
           float* __restrict__ out) {        // [B, N, FOUT]
    __shared__ float s_agg[BATCH * APAD];
    __shared__ int   s_idx[TPB];
    __shared__ float s_w[TPB];
    __shared__ int   s_cnt;

    const int n = blockIdx.x;
    const int t = threadIdx.x;

    // each thread owns 8 elements of the (b,f) = 16x128 tile: flat = t + i*256
    float acc[8];
#pragma unroll
    for (int i = 0; i < 8; ++i) acc[i] = 0.0f;

    const float* adjRow = adj + (size_t)n * N_NODES;

    // scan the adjacency row in 256-wide chunks, compact nonzeros to LDS, gather x
    for (int base = 0; base < N_NODES; base += TPB) {
        if (t == 0) s_cnt = 0;
        __syncthreads();
        float a = adjRow[base + t];
        if (a != 0.0f) {
            int p = atomicAdd(&s_cnt, 1);
            s_idx[p] = base + t;
            s_w[p]   = a * dinv[base + t];
        }
        __syncthreads();
        const int cnt = s_cnt;
        for (int j = 0; j < cnt; ++j) {
            const int   m = s_idx[j];
            const float w = s_w[j];
#pragma unroll
            for (int i = 0; i < 8; ++i) {
                int flat = t + i * TPB;          // b*FEAT + f
                int b = flat >> 7, f = flat & 127;
                acc[i] += w * x[((size_t)b * N_NODES + m) * FEAT + f];
            }
        }
        __syncthreads();
    }

    // self-loop term + D^{-1/2} row scaling; park agg tile (16x128) in padded LDS
    const float dn = dinv[n];
#pragma unroll
    for (int i = 0; i < 8; ++i) {
        int flat = t + i * TPB;
        int b = flat >> 7, f = flat & 127;
        float v = dn * (acc[i] + dn * x[((size_t)b * N_NODES + n) * FEAT + f]);
        s_agg[b * APAD + f] = v;
    }
    __syncthreads();

    // FC: (16 x 128) @ W^T (128 x 128) via V_WMMA_F32_16X16X4_F32.
    // 8 waves, each owns one 16x16 output tile; 32 K-steps of 4. EXEC all-ones here.
    const int lane = t & 31;
    const int wave = t >> 5;          // output-column tile
    const int half = lane & 15;
    const int hi   = lane >> 4;       // ISA f32 A/B layout: K {0,1} in lanes 0-15, {2,3} in 16-31
    const int oBase = wave * 16;

    const float* aPtr = s_agg + half * APAD + hi * 2;                 // A[M=half][K..]
    const float* wPtr = W + (size_t)(oBase + half) * FEAT + hi * 2;   // B[K][o=half] = W[o][K]

    v8f c = {};
#pragma unroll 8
    for (int k = 0; k < FEAT; k += 4) {
        v2f aF = *(const v2f*)(aPtr + k);
        v2f bF = *(const v2f*)(wPtr + k);
        // (neg_a, A, neg_b, B, c_mod, C, reuse_a, reuse_b)
        c = __builtin_amdgcn_wmma_f32_16x16x4_f32(false, aF, false, bF,
                                                  (short)0, c, false, false);
    }

    // D layout: VGPR j -> M = j (lanes 0-15) / j+8 (lanes 16-31), col = lane%16
    const float bb = bias[oBase + half];
#pragma unroll
    for (int j = 0; j < 8; ++j) {
        int b = j + hi * 8;
        out[((size_t)b * N_NODES + n) * FEAT + oBase + half] = c[j] + bb;
    }
}

extern "C" void kernel_launch(void* const* d_in, const int* in_sizes, int n_in,
                              void* d_out, int out_size, void* d_ws, size_t ws_size,
                              hipStream_t stream) {
    const float* adj  = (const float*)d_in[0];   // [N, N]
    const float* x    = (const float*)d_in[1];   // [B, N, FIN]
    const float* W    = (const float*)d_in[2];   // [FOUT, FIN]
    const float* bias = (const float*)d_in[3];   // [FOUT]
    float* out  = (float*)d_out;

    float* deg  = (float*)d_ws;                  // N floats
    float* dinv = deg + N_NODES;                 // N floats  (ws usage: 32 KB)

    gcn_deg_init<<<N_NODES / TPB, TPB, 0, stream>>>(deg);
    gcn_colsum<<<dim3(N_NODES / TPB, 16), TPB, 0, stream>>>(adj, deg);
    gcn_dinv<<<N_NODES / TPB, TPB, 0, stream>>>(deg, dinv);
    gcn_agg_fc<<<N_NODES, TPB, 0, stream>>>(adj, x, W, bias, dinv, out);
}